// CRF_13434657702195
// MI455X (gfx1250) — compile-verified
//
#include <hip/hip_runtime.h>
#include <stdint.h>

#ifndef __has_builtin
#define __has_builtin(x) 0
#endif

#define BB 256
#define TT 512
#define KK 128
#define CHUNK 8
#define NCHUNK (((TT - 1) + CHUNK - 1) / CHUNK)   // 64 chunks cover t = 1..511

typedef int v4i __attribute__((ext_vector_type(4)));
typedef __attribute__((address_space(1))) v4i g_v4i;   // global int4
typedef __attribute__((address_space(3))) v4i l_v4i;   // LDS int4

// ---- CDNA5 async global->LDS copy (ASYNCcnt-tracked) -----------------------
__device__ __forceinline__ void async_load16(const float* g, float* l) {
#if __has_builtin(__builtin_amdgcn_global_load_async_to_lds_b128)
  __builtin_amdgcn_global_load_async_to_lds_b128(
      (g_v4i*)(uintptr_t)g,
      (l_v4i*)(uint32_t)(uintptr_t)l,   // low 32 bits of flat shared ptr == LDS offset
      0, 0);
#else
  uint32_t lo = (uint32_t)(uintptr_t)l;
  asm volatile("global_load_async_to_lds_b128 %0, %1, off"
               :: "v"(lo), "v"((uint64_t)(uintptr_t)g) : "memory");
#endif
}

#if __has_builtin(__builtin_amdgcn_s_wait_asynccnt)
#define WAIT_ASYNC(n) __builtin_amdgcn_s_wait_asynccnt(n)
#else
#define WAIT_ASYNC(n) asm volatile("s_wait_asynccnt %0" :: "n"(n) : "memory")
#endif

// ---------------------------------------------------------------------------
// One block per batch element. 128 threads (4 wave32s). Thread j owns tag j.
//   - transitions (64 KB) resident in LDS, row-major: trans_l[i*K + j]
//   - backpointers for the whole sequence (511*128 u8 = ~64 KB) resident in LDS
//   - potentials streamed via async-to-LDS, double-buffered 8-step chunks
//   - state double-buffered in LDS -> one barrier per timestep
// ---------------------------------------------------------------------------
__global__ __launch_bounds__(KK) void crf_viterbi_onehot(
    const float* __restrict__ potentials,     // (B, T, K) f32
    const float* __restrict__ transitions,    // (K, K)    f32
    const int*   __restrict__ seq_lens,       // (B,)      i32
    float*       __restrict__ out)            // (B, T, K) f32 one-hot
{
  __shared__ __align__(16) float         trans_l[KK * KK];        // 64 KB
  __shared__                unsigned char bp_l[(TT - 1) * KK];    // ~64 KB
  __shared__ __align__(16) float         sbuf[2][KK];             // 1 KB
  __shared__ __align__(16) float         xbuf[2][CHUNK * KK];     // 8 KB
  __shared__                unsigned char tags_l[TT];              // 0.5 KB

  const int b   = blockIdx.x;
  const int j   = threadIdx.x;                 // 0..127 == tag index
  const int len = seq_lens[b];
  const float* potb = potentials + (size_t)b * TT * KK;

  // ---- async-load the whole transition matrix into LDS (32 b128 ops/wave)
  for (int base = 0; base < KK * KK; base += 4 * KK)
    async_load16(transitions + base + j * 4, trans_l + base + j * 4);

  // ---- initial state = potentials[b, 0, :]
  sbuf[0][j] = potb[j];

  // ---- issue chunk 0 (t = 1..8): 1024 floats, 2 x b128 per thread
  {
    const int o0 = j * 4, o1 = o0 + (CHUNK * KK) / 2;
    async_load16(potb + KK + o0, &xbuf[0][o0]);
    async_load16(potb + KK + o1, &xbuf[0][o1]);
  }

  int t = 1;
  for (int c = 0; c < NCHUNK; ++c) {
    // prefetch next chunk while waiting for the current one
    if (c + 1 < NCHUNK) {
      const int base_e = (1 + (c + 1) * CHUNK) * KK;
      const int o0 = j * 4, o1 = o0 + (CHUNK * KK) / 2;
      int g0 = base_e + o0; if (g0 > TT * KK - 4) g0 = TT * KK - 4;  // clamp tail
      int g1 = base_e + o1; if (g1 > TT * KK - 4) g1 = TT * KK - 4;
      float* db = xbuf[(c + 1) & 1];
      async_load16(potb + g0, db + o0);
      async_load16(potb + g1, db + o1);
      WAIT_ASYNC(2);        // chunk c complete for this wave; c+1 in flight
    } else {
      WAIT_ASYNC(0);
    }
    __syncthreads();        // all 4 waves' slices of chunk c are in LDS

    const float* xb    = xbuf[c & 1];
    const int    steps = min(CHUNK, (TT - 1) - c * CHUNK);

    for (int s = 0; s < steps; ++s, ++t) {
      const float* scur = sbuf[(t - 1) & 1];

      // tropical (max,+) GEMV column j with argmax tracking
      float best = -__builtin_inff();
      int   bi   = 0;
      #pragma unroll
      for (int i = 0; i < KK; i += 4) {
        const float4 s4 = *(const float4*)(scur + i);   // LDS broadcast b128
        float v0 = s4.x + trans_l[(i + 0) * KK + j];    // conflict-free b32
        float v1 = s4.y + trans_l[(i + 1) * KK + j];
        float v2 = s4.z + trans_l[(i + 2) * KK + j];
        float v3 = s4.w + trans_l[(i + 3) * KK + j];
        if (v0 > best) { best = v0; bi = i;     }       // first-max tie-break
        if (v1 > best) { best = v1; bi = i + 1; }
        if (v2 > best) { best = v2; bi = i + 2; }
        if (v3 > best) { best = v3; bi = i + 3; }
      }

      float nst;
      if (t < len) {                      // mask: (t-1) < (len-1)
        nst = xb[s * KK + j] + best;
      } else {
        nst = scur[j];                    // frozen state
        bi  = 0;
      }
      bp_l[(t - 1) * KK + j] = (unsigned char)bi;
      sbuf[t & 1][j]         = nst;
      __syncthreads();                    // one barrier per timestep
    }
  }

  // ---- argmax of final score + backpointer chase, entirely in LDS
  if (j == 0) {
    const float* fs = sbuf[(TT - 1) & 1];
    float bmax = fs[0];
    int   tag  = 0;
    for (int i = 1; i < KK; ++i) {
      const float v = fs[i];
      if (v > bmax) { bmax = v; tag = i; }
    }
    tags_l[len - 1] = (unsigned char)tag;
    for (int tt = len - 2; tt >= 0; --tt) {
      tag = bp_l[tt * KK + tag];          // bp[tt] maps tag@tt+1 -> tag@tt
      tags_l[tt] = (unsigned char)tag;
    }
    for (int p = len; p < TT; ++p) tags_l[p] = 0;   // padding decodes to tag 0
  }
  __syncthreads();

  // ---- one-hot expansion, coalesced across lanes
  float* outb = out + (size_t)b * TT * KK;
  for (int tt = 0; tt < TT; ++tt) {
    const int tg = tags_l[tt];
    outb[tt * KK + j] = (j == tg) ? 1.0f : 0.0f;
  }
}

extern "C" void kernel_launch(void* const* d_in, const int* in_sizes, int n_in,
                              void* d_out, int out_size, void* d_ws, size_t ws_size,
                              hipStream_t stream) {
  (void)in_sizes; (void)n_in; (void)out_size; (void)d_ws; (void)ws_size;
  const float* potentials  = (const float*)d_in[0];
  const float* transitions = (const float*)d_in[1];
  const int*   seq_lens    = (const int*)d_in[2];
  float*       out         = (float*)d_out;

  crf_viterbi_onehot<<<dim3(BB), dim3(KK), 0, stream>>>(
      potentials, transitions, seq_lens, out);
}